// GNNLayer_71579924955533
// MI455X (gfx1250) — compile-verified
//
#include <hip/hip_runtime.h>
#include <hip/hip_bf16.h>

typedef __attribute__((ext_vector_type(2))) float v2f;
typedef __attribute__((ext_vector_type(8))) float v8f;

#define IN_D 256
#define OUT_D 64
#define LEAKY 0.01f

// ---------------------------------------------------------------------------
// K1: h = x @ W_lin + b_lin   via V_WMMA_F32_16X16X4_F32
// One wave computes a 16x64 row-tile (4 accumulators of 16x16), K-loop of 64
// steps of K=4. W_lin staged in LDS in K-pair-interleaved layout so every B
// fragment is one aligned ds_load_b64 in exactly the VGPR-pair order the
// WMMA operand wants (no repack movs).
//   sW[(k/2)*128 + c*2 + 0] = W[k][c]
//   sW[(k/2)*128 + c*2 + 1] = W[k+1][c]
// ---------------------------------------------------------------------------
__global__ void __launch_bounds__(256) k_gemm(const float* __restrict__ x,
                                              const float* __restrict__ Wl,
                                              const float* __restrict__ bl,
                                              float* __restrict__ h, int N) {
  __shared__ float sW[IN_D * OUT_D];  // 64 KB, pair-interleaved
  for (int i = threadIdx.x; i < (IN_D / 2) * OUT_D; i += 256) {
    const int kp = i >> 6;   // K pair index
    const int cc = i & 63;   // column
    v2f w = {Wl[(2 * kp) * OUT_D + cc], Wl[(2 * kp + 1) * OUT_D + cc]};
    *(v2f*)(sW + (size_t)i * 2) = w;
  }
  __syncthreads();

  const int wave = threadIdx.x >> 5;
  const int lane = threadIdx.x & 31;
  const int m0 = (blockIdx.x * 8 + wave) * 16;
  if (m0 >= N) return;  // wave-uniform exit, EXEC stays all-ones for WMMA

  // A fragment layout (ISA 7.12.2, 32-bit A 16x4):
  //   lanes 0-15:  M = m0+lane,    VGPR0/1 = K, K+1
  //   lanes 16-31: M = m0+lane-16, VGPR0/1 = K+2, K+3
  int row = m0 + (lane & 15);
  if (row >= N) row = N - 1;          // clamp (select, not branch)
  const int klane = (lane >> 4) * 2;  // 0 or 2
  const int c = lane & 15;
  const float* xrow = x + (size_t)row * IN_D + klane;
  // per-lane LDS base: pair (klane/2), column c -> B pair {W[kk][c],W[kk+1][c]}
  const float* wlane = sW + (klane >> 1) * (OUT_D * 2) + c * 2;

  v8f acc0 = {}, acc1 = {}, acc2 = {}, acc3 = {};
#pragma unroll 4
  for (int k = 0; k < IN_D; k += 4) {
    v2f a = *(const v2f*)(xrow + k);
    const float* wk = wlane + (k >> 1) * (OUT_D * 2);
    v2f b0 = *(const v2f*)(wk);        // cols c      (n-tile 0)
    v2f b1 = *(const v2f*)(wk + 32);   // cols c + 16 (n-tile 1)
    v2f b2 = *(const v2f*)(wk + 64);   // cols c + 32 (n-tile 2)
    v2f b3 = *(const v2f*)(wk + 96);   // cols c + 48 (n-tile 3)
    acc0 = __builtin_amdgcn_wmma_f32_16x16x4_f32(false, a, false, b0, (short)0, acc0, false, false);
    acc1 = __builtin_amdgcn_wmma_f32_16x16x4_f32(false, a, false, b1, (short)0, acc1, false, false);
    acc2 = __builtin_amdgcn_wmma_f32_16x16x4_f32(false, a, false, b2, (short)0, acc2, false, false);
    acc3 = __builtin_amdgcn_wmma_f32_16x16x4_f32(false, a, false, b3, (short)0, acc3, false, false);
  }

  // C/D layout: VGPR i -> M = m0 + i (lanes 0-15) or m0 + 8 + i (lanes 16-31),
  // N = (lane&15) + 16*t. Bias is per-column -> one scalar per accumulator.
  const float bias0 = bl[c];
  const float bias1 = bl[c + 16];
  const float bias2 = bl[c + 32];
  const float bias3 = bl[c + 48];
  const int rbase = m0 + ((lane >> 4) ? 8 : 0);
#pragma unroll
  for (int i = 0; i < 8; ++i) {
    const int r = rbase + i;
    if (r < N) {
      float* hp = h + (size_t)r * OUT_D + c;
      hp[0]  = acc0[i] + bias0;
      hp[16] = acc1[i] + bias1;
      hp[32] = acc2[i] + bias2;
      hp[48] = acc3[i] + bias3;
    }
  }
}

// ---------------------------------------------------------------------------
// K2: per-node attention projections a_src = h.W_att[:64], a_dst = h.W_att[64:]
//     (thread 0 also zeroes the softmax sum accumulator)
// ---------------------------------------------------------------------------
__global__ void k_att(const float* __restrict__ h, const float* __restrict__ Wa,
                      float* __restrict__ asrc, float* __restrict__ adst,
                      float* __restrict__ sum, int N) {
  const int n = blockIdx.x * blockDim.x + threadIdx.x;
  if (n == 0) *sum = 0.0f;
  if (n >= N) return;
  const float* hp = h + (size_t)n * OUT_D;
  float s = 0.f, d = 0.f;
#pragma unroll 8
  for (int k = 0; k < OUT_D; ++k) {
    const float v = hp[k];
    s += v * Wa[k];
    d += v * Wa[OUT_D + k];
  }
  asrc[n] = s;
  adst[n] = d;
}

// ---------------------------------------------------------------------------
// K3: per-edge exp(leaky_relu(score)) + global sum (block reduce + 1 atomic)
// ---------------------------------------------------------------------------
__global__ void __launch_bounds__(256) k_scores(const long long* __restrict__ ei,
                                                const float* __restrict__ asrc,
                                                const float* __restrict__ adst,
                                                const float* __restrict__ ba,
                                                float* __restrict__ esc,
                                                float* __restrict__ sum, int E) {
  const int e = blockIdx.x * 256 + threadIdx.x;
  float v = 0.0f;
  if (e < E) {
    const int r = (int)ei[e];
    const int cidx = (int)ei[(size_t)E + e];
    float s = asrc[r] + adst[cidx] + ba[0];
    s = (s > 0.0f) ? s : LEAKY * s;
    v = expf(s);  // scores ~N(0,1): no max-shift needed for f32 range
    esc[e] = v;
  }
  __shared__ float red[256];
  red[threadIdx.x] = v;
  __syncthreads();
  for (int off = 128; off > 0; off >>= 1) {
    if (threadIdx.x < off) red[threadIdx.x] += red[threadIdx.x + off];
    __syncthreads();
  }
  if (threadIdx.x == 0) atomicAdd(sum, red[0]);
}

// ---------------------------------------------------------------------------
// K4: out <- h (scatter then accumulates directly into d_out)
// ---------------------------------------------------------------------------
__global__ void k_copy(const float* __restrict__ h, float* __restrict__ out, int total4) {
  const int i = blockIdx.x * blockDim.x + threadIdx.x;
  if (i < total4) ((float4*)out)[i] = ((const float4*)h)[i];
}

// ---------------------------------------------------------------------------
// K5: one edge per wave; out[row] += (esc*ew/sum) * h[col]  (64 f32 atomics,
//     2 per lane). h and out rows are L2-resident (25.6 MB each, 192 MB L2).
// ---------------------------------------------------------------------------
__global__ void __launch_bounds__(256) k_scatter(const long long* __restrict__ ei,
                                                 const float* __restrict__ ew,
                                                 const float* __restrict__ esc,
                                                 const float* __restrict__ sum,
                                                 const float* __restrict__ h,
                                                 float* __restrict__ out, int E) {
  const int e = blockIdx.x * 8 + (threadIdx.x >> 5);
  const int lane = threadIdx.x & 31;
  if (e >= E) return;
  const int r = (int)ei[e];
  const int cidx = (int)ei[(size_t)E + e];
  const float s = esc[e] * ew[e] / sum[0];
  const float* hc = h + (size_t)cidx * OUT_D;
  float* op = out + (size_t)r * OUT_D;
  atomicAdd(&op[lane], s * hc[lane]);
  atomicAdd(&op[lane + 32], s * hc[lane + 32]);
}

// ---------------------------------------------------------------------------
// K6: in-place ReLU on out
// ---------------------------------------------------------------------------
__global__ void k_relu(float* __restrict__ out, int total4) {
  const int i = blockIdx.x * blockDim.x + threadIdx.x;
  if (i < total4) {
    float4 v = ((float4*)out)[i];
    v.x = fmaxf(v.x, 0.f);
    v.y = fmaxf(v.y, 0.f);
    v.z = fmaxf(v.z, 0.f);
    v.w = fmaxf(v.w, 0.f);
    ((float4*)out)[i] = v;
  }
}

extern "C" void kernel_launch(void* const* d_in, const int* in_sizes, int n_in,
                              void* d_out, int out_size, void* d_ws, size_t ws_size,
                              hipStream_t stream) {
  const float*     x  = (const float*)d_in[0];
  const long long* ei = (const long long*)d_in[1];  // int64 [2, E]
  const float*     ew = (const float*)d_in[2];
  const float*     Wl = (const float*)d_in[3];
  const float*     bl = (const float*)d_in[4];
  const float*     Wa = (const float*)d_in[5];
  const float*     ba = (const float*)d_in[6];

  const int N = in_sizes[0] / IN_D;  // 100000
  const int E = in_sizes[2];         // 1000000

  // workspace layout (floats): h[N*64] | a_src[N] | a_dst[N] | esc[E] | sum[1]
  float* h    = (float*)d_ws;
  float* asrc = h + (size_t)N * OUT_D;
  float* adst = asrc + N;
  float* esc  = adst + N;
  float* sum  = esc + E;
  float* out  = (float*)d_out;

  k_gemm<<<(N + 127) / 128, 256, 0, stream>>>(x, Wl, bl, h, N);
  k_att<<<(N + 255) / 256, 256, 0, stream>>>(h, Wa, asrc, adst, sum, N);
  k_scores<<<(E + 255) / 256, 256, 0, stream>>>(ei, asrc, adst, ba, esc, sum, E);
  const int total4 = N * (OUT_D / 4);
  k_copy<<<(total4 + 255) / 256, 256, 0, stream>>>(h, out, total4);
  k_scatter<<<(E + 7) / 8, 256, 0, stream>>>(ei, ew, esc, sum, h, out, E);
  k_relu<<<(total4 + 255) / 256, 256, 0, stream>>>(out, total4);
}